// MassSprings_7378753815243
// MI455X (gfx1250) — compile-verified
//
#include <hip/hip_runtime.h>

typedef float v2f __attribute__((ext_vector_type(2)));
typedef float v8f __attribute__((ext_vector_type(8)));
typedef float vf4 __attribute__((ext_vector_type(4)));
typedef int   vi4 __attribute__((ext_vector_type(4)));

#define MS_EPS 1e-6f
#define MS_NBLOCKS 1024
#define MS_TPB 256

// Full 32-lane sum via V_WMMA_F32_16X16X4_F32.
// A[16x4] striped per ISA layout: lane L<16 holds A[L][0],A[L][1]; lane L>=16
// holds A[L-16][2],A[L-16][3]. With B = all ones, D[m][n] = sum_k A[m][k]
// = p_m + p_{m+16}. D VGPR r: lanes 0-15 -> M=r, lanes 16-31 -> M=8+r, so
// summing the 8 D regs per lane gives half-sum; one xor-16 shuffle finishes.
// EXEC must be all ones at the call site (post-loop convergence guarantees it).
__device__ __forceinline__ float wave_sum32(float p) {
    v2f a; a[0] = p;    a[1] = 0.0f;
    v2f b; b[0] = 1.0f; b[1] = 1.0f;
    v8f c = {};
    v8f d = __builtin_amdgcn_wmma_f32_16x16x4_f32(
        /*neg_a=*/false, a, /*neg_b=*/false, b,
        /*c_mod=*/(short)0, c, /*reuse_a=*/false, /*reuse_b=*/false);
    float s = d[0] + d[1] + d[2] + d[3] + d[4] + d[5] + d[6] + d[7];
    s += __shfl_xor(s, 16, 32);
    return s;  // every lane holds the full 32-lane sum
}

__device__ __forceinline__ float spring_e(const float* __restrict__ x,
                                          int i1, int i2, float l0, float k) {
    // x lives in L2 (1.2 MB << 192 MB); gathers are regular cached loads.
    float dx = x[3 * i1 + 0] - x[3 * i2 + 0];
    float dy = x[3 * i1 + 1] - x[3 * i2 + 1];
    float dz = x[3 * i1 + 2] - x[3 * i2 + 2];
    float q  = fmaf(dx, dx, fmaf(dy, dy, dz * dz));
    float l  = sqrtf(q + MS_EPS);
    float dl = l - l0;
    return k * dl * dl;
}

__global__ __launch_bounds__(MS_TPB)
void ms_partial_kernel(const float* __restrict__ x,
                       const float* __restrict__ l0,
                       const float* __restrict__ k,
                       const int*   __restrict__ idx,
                       float*       __restrict__ partials,
                       int E) {
    const int tid      = blockIdx.x * blockDim.x + threadIdx.x;
    const int nthreads = gridDim.x * blockDim.x;
    const int E4       = E >> 2;

    float acc = 0.0f;
    // 4 springs per iteration: float4 l0/k + 2x int4 indices, all B128
    // non-temporal loads (don't let the 103 MB stream evict the hot x table).
    for (int g = tid; g < E4; g += nthreads) {
        vf4 l0v = __builtin_nontemporal_load((const vf4*)l0 + g);
        vf4 kv  = __builtin_nontemporal_load((const vf4*)k  + g);
        vi4 iA  = __builtin_nontemporal_load((const vi4*)idx + 2 * g);
        vi4 iB  = __builtin_nontemporal_load((const vi4*)idx + 2 * g + 1);
        acc += spring_e(x, iA[0], iA[1], l0v[0], kv[0]);
        acc += spring_e(x, iA[2], iA[3], l0v[1], kv[1]);
        acc += spring_e(x, iB[0], iB[1], l0v[2], kv[2]);
        acc += spring_e(x, iB[2], iB[3], l0v[3], kv[3]);
    }
    // Scalar tail (empty when E % 4 == 0, as here).
    for (int e = (E4 << 2) + tid; e < E; e += nthreads) {
        acc += spring_e(x, idx[2 * e + 0], idx[2 * e + 1], l0[e], k[e]);
    }

    // Deterministic block reduction: WMMA wave sum -> LDS -> thread 0.
    __shared__ float wsum[MS_TPB / 32];
    float w = wave_sum32(acc);
    const int lane = threadIdx.x & 31;
    const int wid  = threadIdx.x >> 5;
    if (lane == 0) wsum[wid] = w;
    __syncthreads();
    if (threadIdx.x == 0) {
        float t = 0.0f;
#pragma unroll
        for (int i = 0; i < MS_TPB / 32; ++i) t += wsum[i];
        partials[blockIdx.x] = t;
    }
}

__global__ __launch_bounds__(MS_TPB)
void ms_final_kernel(const float* __restrict__ partials, int n,
                     float* __restrict__ out) {
    float acc = 0.0f;
    for (int i = threadIdx.x; i < n; i += blockDim.x) acc += partials[i];

    __shared__ float wsum[MS_TPB / 32];
    float w = wave_sum32(acc);
    const int lane = threadIdx.x & 31;
    const int wid  = threadIdx.x >> 5;
    if (lane == 0) wsum[wid] = w;
    __syncthreads();
    if (threadIdx.x == 0) {
        float t = 0.0f;
#pragma unroll
        for (int i = 0; i < MS_TPB / 32; ++i) t += wsum[i];
        out[0] = 0.5f * t;  // E = 0.5 * sum(k * dl^2)
    }
}

extern "C" void kernel_launch(void* const* d_in, const int* in_sizes, int n_in,
                              void* d_out, int out_size, void* d_ws, size_t ws_size,
                              hipStream_t stream) {
    // setup_inputs order: x [V*3] f32, l0 [E] f32, k [E] f32, indices [E*2] i32
    const float* x   = (const float*)d_in[0];
    const float* l0  = (const float*)d_in[1];
    const float* k   = (const float*)d_in[2];
    const int*   idx = (const int*)d_in[3];
    const int E = in_sizes[1];

    float* partials = (float*)d_ws;  // MS_NBLOCKS floats of scratch
    float* out      = (float*)d_out;

    ms_partial_kernel<<<MS_NBLOCKS, MS_TPB, 0, stream>>>(x, l0, k, idx, partials, E);
    ms_final_kernel<<<1, MS_TPB, 0, stream>>>(partials, MS_NBLOCKS, out);
}